// RepulsionLoss_7447473291842
// MI455X (gfx1250) — compile-verified
//
#include <hip/hip_runtime.h>

typedef __attribute__((ext_vector_type(2))) float v2f;
typedef __attribute__((ext_vector_type(4))) float v4f;
typedef __attribute__((ext_vector_type(8))) float v8f;

#define BATCH 32
#define NPTS 1024
#define DIM 16
#define KNN 5
#define ITILES (NPTS / 16)                 // 64 tiles of 16 rows
#define TOTAL_TILES (BATCH * ITILES)       // 2048
#define WAVES_PER_BLOCK 8
#define BIGF 3.0e38f

// Branchless sorted-insert keeping the 5 smallest (t[0..4] ascending).
__device__ __forceinline__ void ins5(float v, float* t) {
#pragma unroll
  for (int i = 0; i < 5; ++i) {
    float lo = fminf(v, t[i]);
    float hi = fmaxf(v, t[i]);
    t[i] = lo;
    v = hi;
  }
}

// Kernel 1: per-point squared norms (32768 points, 16 dims each).
__global__ void __launch_bounds__(256)
sq_kernel(const float* __restrict__ X, float* __restrict__ sq) {
  int p = blockIdx.x * blockDim.x + threadIdx.x;
  const v4f* row = (const v4f*)(X + (size_t)p * DIM);
  float s = 0.f;
#pragma unroll
  for (int i = 0; i < 4; ++i) {
    v4f r = row[i];
    s += r.x * r.x + r.y * r.y + r.z * r.z + r.w * r.w;
  }
  sq[p] = s;
}

// Kernel 2: fused Gram (f32 WMMA) + distance + running top-5 + repulsion.
// One wave32 per 16-row i-tile; loops over all 64 j-tiles of its batch.
__global__ void __launch_bounds__(256)
repulsion_kernel(const float* __restrict__ X, const float* __restrict__ sq,
                 float* __restrict__ partials) {
  __shared__ float cand[WAVES_PER_BLOCK][16 * 16 * KNN];  // [wave][row][srcLane][k]
  __shared__ float rowsum[WAVES_PER_BLOCK][16];

  const int lane = threadIdx.x & 31;
  const int wave = threadIdx.x >> 5;
  const int tile = blockIdx.x * WAVES_PER_BLOCK + wave;   // 0..2047
  const int b  = tile / ITILES;
  const int it = tile % ITILES;

  const int col = lane & 15;   // A-row / B-col owned by this lane
  const int h   = lane >> 4;   // lane-half selects K-pair (A) / row-set (C)

  const float* Xb  = X + (size_t)b * NPTS * DIM;
  const float* sqb = sq + b * NPTS;

  // A operand: row (col), K-pair (2h) within each K=4 step -> aligned float2.
  const float* xa = Xb + (it * 16 + col) * DIM + 2 * h;
  v2f a0 = *(const v2f*)(xa + 0);
  v2f a1 = *(const v2f*)(xa + 4);
  v2f a2 = *(const v2f*)(xa + 8);
  v2f a3 = *(const v2f*)(xa + 12);

  // Squared norms of the 8 C-rows this lane holds (rows v+8h, v=0..7).
  float sqi[8];
#pragma unroll
  for (int v = 0; v < 8; ++v) sqi[v] = sqb[it * 16 + v + 8 * h];

  float top[8][KNN];
#pragma unroll
  for (int v = 0; v < 8; ++v)
#pragma unroll
    for (int k = 0; k < KNN; ++k) top[v][k] = BIGF;

  for (int jt = 0; jt < ITILES; ++jt) {
    const float* xr = Xb + (jt * 16 + col) * DIM + 2 * h;
    v2f b0 = *(const v2f*)(xr + 0);
    v2f b1 = *(const v2f*)(xr + 4);
    v2f b2 = *(const v2f*)(xr + 8);
    v2f b3 = *(const v2f*)(xr + 12);
    float sqj = sqb[jt * 16 + col];

    // Full-precision Gram tile: 4 chained V_WMMA_F32_16X16X4_F32 cover K=16.
    v8f c = {0.f, 0.f, 0.f, 0.f, 0.f, 0.f, 0.f, 0.f};
    c = __builtin_amdgcn_wmma_f32_16x16x4_f32(false, a0, false, b0, (short)0, c, false, false);
    c = __builtin_amdgcn_wmma_f32_16x16x4_f32(false, a1, false, b1, (short)0, c, false, false);
    c = __builtin_amdgcn_wmma_f32_16x16x4_f32(false, a2, false, b2, (short)0, c, false, false);
    c = __builtin_amdgcn_wmma_f32_16x16x4_f32(false, a3, false, b3, (short)0, c, false, false);

#pragma unroll
    for (int v = 0; v < 8; ++v) {
      int row = v + 8 * h;                         // C layout: VGPR v -> M = v + 8*half
      float d = sqi[v] + sqj - 2.0f * c[v];
      d = fmaxf(d, 0.0f);
      bool diag = (it == jt) && (row == col);      // self-distance -> never selected
      d = diag ? BIGF : d;
      ins5(d, top[v]);
    }
  }

  // Merge per-row top-5 across the 16 lanes that share each row.
#pragma unroll
  for (int v = 0; v < 8; ++v) {
    int row = v + 8 * h;
#pragma unroll
    for (int k = 0; k < KNN; ++k)
      cand[wave][(row * 16 + col) * KNN + k] = top[v][k];
  }
  __syncthreads();

  if (lane < 16) {
    float m[KNN];
#pragma unroll
    for (int k = 0; k < KNN; ++k) m[k] = BIGF;
    for (int c2 = 0; c2 < 16; ++c2)
#pragma unroll
      for (int k = 0; k < KNN; ++k)
        ins5(cand[wave][(lane * 16 + c2) * KNN + k], m);
    float s = 0.f;
#pragma unroll
    for (int k = 0; k < KNN; ++k)
      s += 1.0f / (0.01f + m[k]);                  // LAMBDA=1, DELTA=0.01, S/2=1
    rowsum[wave][lane] = s;
  }
  __syncthreads();

  if (lane == 0) {
    float s = 0.f;
#pragma unroll
    for (int r = 0; r < 16; ++r) s += rowsum[wave][r];
    partials[tile] = s;
  }
}

// Kernel 3: deterministic reduction of 2048 tile partials -> mean scalar.
__global__ void __launch_bounds__(256)
reduce_kernel(const float* __restrict__ partials, float* __restrict__ out) {
  __shared__ float buf[256];
  float s = 0.f;
  for (int i = threadIdx.x; i < TOTAL_TILES; i += 256) s += partials[i];
  buf[threadIdx.x] = s;
  __syncthreads();
  for (int st = 128; st > 0; st >>= 1) {
    if (threadIdx.x < st) buf[threadIdx.x] += buf[threadIdx.x + st];
    __syncthreads();
  }
  if (threadIdx.x == 0)
    out[0] = buf[0] * (1.0f / ((float)BATCH * (float)NPTS * (float)KNN));
}

extern "C" void kernel_launch(void* const* d_in, const int* in_sizes, int n_in,
                              void* d_out, int out_size, void* d_ws, size_t ws_size,
                              hipStream_t stream) {
  (void)in_sizes; (void)n_in; (void)out_size; (void)ws_size;
  const float* X = (const float*)d_in[0];
  float* sq = (float*)d_ws;                 // BATCH*NPTS floats (128 KB)
  float* partials = sq + BATCH * NPTS;      // TOTAL_TILES floats (8 KB)

  sq_kernel<<<(BATCH * NPTS) / 256, 256, 0, stream>>>(X, sq);
  repulsion_kernel<<<TOTAL_TILES / WAVES_PER_BLOCK, 256, 0, stream>>>(X, sq, partials);
  reduce_kernel<<<1, 256, 0, stream>>>(partials, (float*)d_out);
}